// SMPL2MeshOptimizer_29240137351258
// MI455X (gfx1250) — compile-verified
//
#include <hip/hip_runtime.h>

typedef __attribute__((ext_vector_type(2))) float v2f;
typedef __attribute__((ext_vector_type(8))) float v8f;

#define EPS_VN  1e-6f
#define EPS_COS 1e-8f

// Accumulator slots (in workspace):
// 0: chamfer total (pre-weighted, pre-mean-scaled per direction)
// 1: edge-loss raw sum       (-> *0.1 / F)
// 2: smooth raw sum          (-> *0.1 / (3F))
// 3: normal-consistency sum  (-> *0.1 / E)
// 4: laplacian raw sum       (-> *0.1 / V)
// 5: normal-delta sq sum     (-> *0.1 / (3V))
// 6: penetration raw sum     (-> *0.1 / V)

__device__ inline float waveReduceSum(float v) {
    v += __shfl_xor(v, 16, 32);
    v += __shfl_xor(v, 8, 32);
    v += __shfl_xor(v, 4, 32);
    v += __shfl_xor(v, 2, 32);
    v += __shfl_xor(v, 1, 32);
    return v;
}

// -------------------------------------------------------------------------
// Init: new_v = verts + deform, flip_v, squared norms, zero accumulators.
// -------------------------------------------------------------------------
__global__ __launch_bounds__(256)
void k_init(const float* __restrict__ verts, const float* __restrict__ deform,
            const float* __restrict__ trg, int V,
            float* __restrict__ newv, float* __restrict__ flipv,
            float* __restrict__ initn, float* __restrict__ newn,
            float* __restrict__ Lv, float* __restrict__ rs,
            float* __restrict__ nnew, float* __restrict__ ntrg,
            float* __restrict__ acc)
{
    int i = blockIdx.x * blockDim.x + threadIdx.x;
    if (i < 16) acc[i] = 0.0f;
    if (i >= V) return;
    float vx = verts[3*i+0], vy = verts[3*i+1], vz = verts[3*i+2];
    float dx = deform[3*i+0], dy = deform[3*i+1], dz = deform[3*i+2];
    float nx = vx + dx, ny = vy + dy, nz = vz + dz;
    newv[3*i+0] = nx;  newv[3*i+1] = ny;  newv[3*i+2] = nz;
    flipv[3*i+0] = -nx; flipv[3*i+1] = ny; flipv[3*i+2] = nz;
    nnew[i] = nx*nx + ny*ny + nz*nz;
    float tx = trg[3*i+0], ty = trg[3*i+1], tz = trg[3*i+2];
    ntrg[i] = tx*tx + ty*ty + tz*tz;
    initn[3*i+0] = 0.f; initn[3*i+1] = 0.f; initn[3*i+2] = 0.f;
    newn[3*i+0]  = 0.f; newn[3*i+1]  = 0.f; newn[3*i+2]  = 0.f;
    Lv[3*i+0]    = 0.f; Lv[3*i+1]    = 0.f; Lv[3*i+2]    = 0.f;
    rs[i] = 0.f;
}

// -------------------------------------------------------------------------
// Directional chamfer via V_WMMA_F32_16X16X4_F32, M-blocked 4x, N-unrolled 2x.
//   adds  scale * sum_i min_j ||x_i - y_j||^2   into *accum.
// Each wave owns 4 row tiles (64 rows of X); each trip processes 2 B tiles
// (8 WMMAs), with the two candidates folded via fminf(fminf) -> v_min3_num.
// A (16x4 f32, 2 VGPRs): lanes 0-15 hold M=lane {K0,K1}, lanes 16-31 {K2,K3}.
// B (4x16 f32, 2 VGPRs): lanes 0-15 hold N=lane {K0,K1}, lanes 16-31 {K2,K3}.
// C (16x16 f32, 8 VGPRs): VGPR r, lanes 0-15 -> (M=r, N=lane),
//                                  lanes 16-31 -> (M=8+r, N=lane-16).
// K = {x, y, z, 0}.
// -------------------------------------------------------------------------
#define MTILES 4

__global__ __launch_bounds__(256)
void k_chamfer_dir(const float* __restrict__ X, const float* __restrict__ Xn,
                   const float* __restrict__ Y, const float* __restrict__ Yn,
                   int nX, int nY, float scale, float* __restrict__ accum)
{
    const int lane = threadIdx.x & 31;
    const int wave = threadIdx.x >> 5;
    const int nTilesX = (nX + 15) >> 4;
    const int nTasks  = (nTilesX + MTILES - 1) / MTILES;
    const int task    = blockIdx.x * 8 + wave;   // wave-uniform
    if (task >= nTasks) return;                  // EXEC stays all-1 for WMMA
    const int rt0 = task * MTILES;               // first 16-row tile of this wave

    const int half = lane >> 4;                  // 0: K0/K1 lanes, 1: K2/K3 lanes
    const int l16  = lane & 15;

    // ---- load 4 A strips (branch-free coordinate select) ----
    v2f A[MTILES];
#pragma unroll
    for (int t = 0; t < MTILES; ++t) {
        const int m = (rt0 + t) * 16 + l16;
        float x0 = 0.f, x1 = 0.f;
        if (m < nX) {
            x0 = X[3*m + 2*half];                // K0 (x) or K2 (z)
            float xy = X[3*m + 1];
            x1 = (half == 0) ? xy : 0.f;         // K1 (y) or K3 (pad 0)
        }
        A[t].x = x0; A[t].y = x1;
    }

    float mins[MTILES][8];
#pragma unroll
    for (int t = 0; t < MTILES; ++t)
#pragma unroll
        for (int r = 0; r < 8; ++r) mins[t][r] = 3.0e38f;

    // ---- guard-free sweep, 2 column tiles per trip ----
    const int nFull = nY >> 4;
    int ct = 0;
    for (; ct + 2 <= nFull; ct += 2) {
        const int n0 = ct * 16 + l16;            // matches C-layout column N
        const int n1 = n0 + 16;
        if (ct + 16 < nFull)                     // global_prefetch_b8 path
            __builtin_prefetch(&Y[3 * ((ct + 16) * 16 + l16)], 0, 1);
        // issue both B tiles' loads up front
        float b00 = Y[3*n0 + 2*half];
        float b0y = Y[3*n0 + 1];
        float yy0 = Yn[n0];
        float b10 = Y[3*n1 + 2*half];
        float b1y = Y[3*n1 + 1];
        float yy1 = Yn[n1];
        v2f B0; B0.x = b00; B0.y = (half == 0) ? b0y : 0.f;
        v2f B1; B1.x = b10; B1.y = (half == 0) ? b1y : 0.f;

#pragma unroll
        for (int t = 0; t < MTILES; ++t) {
            v8f c0 = {}, c1 = {};
            c0 = __builtin_amdgcn_wmma_f32_16x16x4_f32(
                false, A[t], false, B0, (short)0, c0, false, false);
            c1 = __builtin_amdgcn_wmma_f32_16x16x4_f32(
                false, A[t], false, B1, (short)0, c1, false, false);
#pragma unroll
            for (int r = 0; r < 8; ++r) {
                float d0 = yy0 - 2.0f * c0[r];
                float d1 = yy1 - 2.0f * c1[r];
                mins[t][r] = fminf(mins[t][r], fminf(d0, d1));  // -> v_min3_num
            }
        }
    }

    // ---- odd full tile (when nFull is odd) ----
    if (ct < nFull) {
        const int n = ct * 16 + l16;
        float b0 = Y[3*n + 2*half];
        float by = Y[3*n + 1];
        float yy = Yn[n];
        v2f B; B.x = b0; B.y = (half == 0) ? by : 0.f;
#pragma unroll
        for (int t = 0; t < MTILES; ++t) {
            v8f c = {};
            c = __builtin_amdgcn_wmma_f32_16x16x4_f32(
                false, A[t], false, B, (short)0, c, false, false);
#pragma unroll
            for (int r = 0; r < 8; ++r)
                mins[t][r] = fminf(mins[t][r], yy - 2.0f * c[r]);
        }
    }

    // ---- guarded tail tile (absent when nY % 16 == 0) ----
    if (nY & 15) {
        const int n = nFull * 16 + l16;
        float b0 = 0.f, b1 = 0.f, yy = 3.0e38f;
        if (n < nY) {
            b0 = Y[3*n + 2*half];
            float by = Y[3*n + 1];
            b1 = (half == 0) ? by : 0.f;
            yy = Yn[n];
        }
        v2f B; B.x = b0; B.y = b1;
#pragma unroll
        for (int t = 0; t < MTILES; ++t) {
            v8f c = {};
            c = __builtin_amdgcn_wmma_f32_16x16x4_f32(
                false, A[t], false, B, (short)0, c, false, false);
#pragma unroll
            for (int r = 0; r < 8; ++r)
                mins[t][r] = fminf(mins[t][r], yy - 2.0f * c[r]);
        }
    }

    // ---- min across the 16 lanes of each half, then accumulate ----
    float total = 0.f;
#pragma unroll
    for (int t = 0; t < MTILES; ++t) {
#pragma unroll
        for (int r = 0; r < 8; ++r) {
            float v = mins[t][r];
            v = fminf(v, __shfl_xor(v, 8, 32));
            v = fminf(v, __shfl_xor(v, 4, 32));
            v = fminf(v, __shfl_xor(v, 2, 32));
            v = fminf(v, __shfl_xor(v, 1, 32));
            // lane 0 -> rows +0..7 of tile t ; lane 16 -> rows +8..15
            int row = (rt0 + t) * 16 + half * 8 + r;
            if (l16 == 0 && row < nX) total += Xn[row] + v;
        }
    }
    if (l16 == 0) atomicAdd(accum, total * scale);
}

// -------------------------------------------------------------------------
// Per-face: face normals (init & new), edge loss, smooth loss, cot-Laplacian
// -------------------------------------------------------------------------
__device__ inline void ld3(const float* p, int i, float v[3]) {
    v[0] = p[3*i+0]; v[1] = p[3*i+1]; v[2] = p[3*i+2];
}
__device__ inline void cross3(const float a[3], const float b[3], float o[3]) {
    o[0] = a[1]*b[2] - a[2]*b[1];
    o[1] = a[2]*b[0] - a[0]*b[2];
    o[2] = a[0]*b[1] - a[1]*b[0];
}
__device__ inline float len3(const float a[3]) {
    return sqrtf(a[0]*a[0] + a[1]*a[1] + a[2]*a[2]);
}

__global__ __launch_bounds__(256)
void k_face(const float* __restrict__ verts, const float* __restrict__ deform,
            const float* __restrict__ newv, const int* __restrict__ faces, int F,
            float* __restrict__ initn, float* __restrict__ newn,
            float* __restrict__ Lv, float* __restrict__ rs,
            float* __restrict__ acc)
{
    int f = blockIdx.x * blockDim.x + threadIdx.x;
    float edge = 0.f, sm = 0.f;
    if (f < F) {
        int ia = faces[3*f+0], ib = faces[3*f+1], ic = faces[3*f+2];
        int idx[3] = {ia, ib, ic};

        // ---- face normals (init from verts, new from newv) ----
        float p0[3], p1[3], p2[3], e1[3], e2[3], n[3];
        ld3(verts, ia, p0); ld3(verts, ib, p1); ld3(verts, ic, p2);
        for (int k = 0; k < 3; ++k) { e1[k] = p1[k]-p0[k]; e2[k] = p2[k]-p0[k]; }
        cross3(e1, e2, n);
        for (int k = 0; k < 3; ++k) {
            atomicAdd(&initn[3*ia+k], n[k]);
            atomicAdd(&initn[3*ib+k], n[k]);
            atomicAdd(&initn[3*ic+k], n[k]);
        }
        float q0[3], q1[3], q2[3];
        ld3(newv, ia, q0); ld3(newv, ib, q1); ld3(newv, ic, q2);
        for (int k = 0; k < 3; ++k) { e1[k] = q1[k]-q0[k]; e2[k] = q2[k]-q0[k]; }
        cross3(e1, e2, n);
        for (int k = 0; k < 3; ++k) {
            atomicAdd(&newn[3*ia+k], n[k]);
            atomicAdd(&newn[3*ib+k], n[k]);
            atomicAdd(&newn[3*ic+k], n[k]);
        }

        // ---- edge loss (new positions) ----
        float d01[3], d12[3], d20[3];
        for (int k = 0; k < 3; ++k) { d01[k]=q0[k]-q1[k]; d12[k]=q1[k]-q2[k]; d20[k]=q2[k]-q0[k]; }
        float e0l = len3(d01), e1l = len3(d12), e2l = len3(d20);
        edge = (e0l-e1l)*(e0l-e1l) + (e1l-e2l)*(e1l-e2l) + (e2l-e0l)*(e2l-e0l);

        // ---- smooth loss: offsets = deform gathered per face ----
        float da[3], db[3], dc[3];
        ld3(deform, ia, da); ld3(deform, ib, db); ld3(deform, ic, dc);
        for (int k = 0; k < 3; ++k)
            sm += fabsf(da[k]-db[k]) + fabsf(db[k]-dc[k]) + fabsf(dc[k]-da[k]);

        // ---- cot-Laplacian weights (new positions) ----
        float la = e1l;        // |v1 - v2|
        float lb = len3(d20);  // |v0 - v2|
        float lc = e0l;        // |v0 - v1|
        float s = 0.5f * (la + lb + lc);
        float area2 = s * (s - la) * (s - lb) * (s - lc);
        float area = sqrtf(fmaxf(area2, 1e-12f));
        float inv4a = 1.0f / (4.0f * area);
        float cotw[3];
        cotw[0] = (lb*lb + lc*lc - la*la) * inv4a;
        cotw[1] = (la*la + lc*lc - lb*lb) * inv4a;
        cotw[2] = (la*la + lb*lb - lc*lc) * inv4a;
        const float* q[3] = {q0, q1, q2};
        for (int k = 0; k < 3; ++k) {
            int i = idx[(k+1)%3], j = idx[(k+2)%3];
            float w = cotw[k];
            for (int c = 0; c < 3; ++c) {
                atomicAdd(&Lv[3*i+c], w * q[(k+2)%3][c]);
                atomicAdd(&Lv[3*j+c], w * q[(k+1)%3][c]);
            }
            atomicAdd(&rs[i], w);
            atomicAdd(&rs[j], w);
        }
    }
    // full-wave reduction (lanes reconverged; invalid lanes contribute 0)
    edge = waveReduceSum(edge);
    sm   = waveReduceSum(sm);
    if ((threadIdx.x & 31) == 0) {
        atomicAdd(&acc[1], edge);
        atomicAdd(&acc[2], sm);
    }
}

// -------------------------------------------------------------------------
// Per interior edge: normal consistency
// -------------------------------------------------------------------------
__global__ __launch_bounds__(256)
void k_edge_nc(const float* __restrict__ newv, const int* __restrict__ ev,
               const int* __restrict__ eo, int E, float* __restrict__ acc)
{
    int e = blockIdx.x * blockDim.x + threadIdx.x;
    float nc = 0.f;
    if (e < E) {
        int a = ev[2*e+0], b = ev[2*e+1];
        int o0 = eo[2*e+0], o1 = eo[2*e+1];
        float pa[3], pb[3], p0[3], p1[3];
        ld3(newv, a, pa); ld3(newv, b, pb); ld3(newv, o0, p0); ld3(newv, o1, p1);
        float ed[3], u0[3], u1[3], n0[3], n1[3];
        for (int k = 0; k < 3; ++k) {
            ed[k] = pb[k]-pa[k]; u0[k] = p0[k]-pa[k]; u1[k] = p1[k]-pa[k];
        }
        cross3(ed, u0, n0);
        cross3(ed, u1, n1);
        for (int k = 0; k < 3; ++k) n1[k] = -n1[k];
        float dot = n0[0]*n1[0] + n0[1]*n1[1] + n0[2]*n1[2];
        float cosv = dot / fmaxf(len3(n0) * len3(n1), EPS_COS);
        nc = 1.0f - cosv;
    }
    nc = waveReduceSum(nc);
    if ((threadIdx.x & 31) == 0) atomicAdd(&acc[3], nc);
}

// -------------------------------------------------------------------------
// Per vertex: normalize normals, consistency, penetration, laplacian finish
// -------------------------------------------------------------------------
__global__ __launch_bounds__(256)
void k_vertex(const float* __restrict__ deform, const float* __restrict__ newv,
              const float* __restrict__ initn, const float* __restrict__ newn,
              const float* __restrict__ Lv, const float* __restrict__ rs,
              int V, float* __restrict__ acc)
{
    int i = blockIdx.x * blockDim.x + threadIdx.x;
    float cons = 0.f, penv = 0.f, lap = 0.f;
    if (i < V) {
        float ni[3], nn[3];
        ld3(initn, i, ni); ld3(newn, i, nn);
        float li = fmaxf(len3(ni), EPS_VN);
        float ln = fmaxf(len3(nn), EPS_VN);
        for (int k = 0; k < 3; ++k) {
            ni[k] /= li; nn[k] /= ln;
            float d = nn[k] - ni[k];
            cons += d * d;
        }
        float dd[3]; ld3(deform, i, dd);
        float pen = dd[0]*ni[0] + dd[1]*ni[1] + dd[2]*ni[2];
        penv = fabsf(fminf(pen, 0.0f));

        float r = rs[i];
        float inv = (r > 0.f) ? (1.0f / r) : 0.0f;
        float lv[3], v[3];
        ld3(Lv, i, lv); ld3(newv, i, v);
        float dl[3];
        for (int k = 0; k < 3; ++k) dl[k] = lv[k] * inv - v[k];
        lap = len3(dl);
    }
    cons = waveReduceSum(cons);
    penv = waveReduceSum(penv);
    lap  = waveReduceSum(lap);
    if ((threadIdx.x & 31) == 0) {
        atomicAdd(&acc[5], cons);
        atomicAdd(&acc[6], penv);
        atomicAdd(&acc[4], lap);
    }
}

// -------------------------------------------------------------------------
// Final combine -> scalar loss
// -------------------------------------------------------------------------
__global__ void k_combine(const float* __restrict__ acc, int V, int F, int E,
                          float* __restrict__ out)
{
    float loss = acc[0];                            // chamfer terms (pre-weighted)
    loss += 0.1f * (acc[1] / (float)F);             // edge
    loss += 0.1f * (acc[2] / (3.0f * (float)F));    // smooth
    loss += 0.1f * (acc[3] / (float)E);             // normal consistency
    loss += 0.1f * (acc[4] / (float)V);             // laplacian
    loss += 0.1f * (acc[5] / (3.0f * (float)V));    // consistency
    loss += 0.1f * (acc[6] / (float)V);             // penetration
    out[0] = loss;
}

// -------------------------------------------------------------------------
extern "C" void kernel_launch(void* const* d_in, const int* in_sizes, int n_in,
                              void* d_out, int out_size, void* d_ws, size_t ws_size,
                              hipStream_t stream)
{
    const float* verts  = (const float*)d_in[0];
    const float* deform = (const float*)d_in[1];
    const float* trg    = (const float*)d_in[2];
    const int*   faces  = (const int*)d_in[3];
    const int*   ev     = (const int*)d_in[4];
    const int*   eo     = (const int*)d_in[5];
    const int V = in_sizes[0] / 3;
    const int F = in_sizes[3] / 3;
    const int E = in_sizes[4] / 2;

    float* w     = (float*)d_ws;
    float* newv  = w;               // 3V
    float* flipv = newv  + 3*V;     // 3V
    float* initn = flipv + 3*V;     // 3V
    float* newn  = initn + 3*V;     // 3V
    float* Lv    = newn  + 3*V;     // 3V
    float* rs    = Lv    + 3*V;     // V
    float* nnew  = rs    + V;       // V
    float* ntrg  = nnew  + V;       // V
    float* acc   = ntrg  + V;       // 16

    const int T = 256;
    k_init<<<(V + T - 1) / T, T, 0, stream>>>(verts, deform, trg, V,
                                              newv, flipv, initn, newn,
                                              Lv, rs, nnew, ntrg, acc);

    const int nTilesX = (V + 15) >> 4;
    const int nTasks  = (nTilesX + MTILES - 1) / MTILES;
    const int cblocks = (nTasks + 7) / 8;      // 8 waves per block
    const float inv_v = 1.0f / (float)V;
    // chamfer(new_v, trg), weight 1.0
    k_chamfer_dir<<<cblocks, T, 0, stream>>>(newv, nnew, trg,  ntrg, V, V, 1.0f * inv_v, &acc[0]);
    k_chamfer_dir<<<cblocks, T, 0, stream>>>(trg,  ntrg, newv, nnew, V, V, 1.0f * inv_v, &acc[0]);
    // chamfer(new_v, flip(new_v)), weight 0.1 (flip preserves norms -> reuse nnew)
    k_chamfer_dir<<<cblocks, T, 0, stream>>>(newv,  nnew, flipv, nnew, V, V, 0.1f * inv_v, &acc[0]);
    k_chamfer_dir<<<cblocks, T, 0, stream>>>(flipv, nnew, newv,  nnew, V, V, 0.1f * inv_v, &acc[0]);

    k_face<<<(F + T - 1) / T, T, 0, stream>>>(verts, deform, newv, faces, F,
                                              initn, newn, Lv, rs, acc);
    k_edge_nc<<<(E + T - 1) / T, T, 0, stream>>>(newv, ev, eo, E, acc);
    k_vertex<<<(V + T - 1) / T, T, 0, stream>>>(deform, newv, initn, newn,
                                                Lv, rs, V, acc);
    k_combine<<<1, 1, 0, stream>>>(acc, V, F, E, (float*)d_out);
}